// DiffPathRenderer_755914244213
// MI455X (gfx1250) — compile-verified
//
#include <hip/hip_runtime.h>
#include <hip/hip_bf16.h>

typedef __attribute__((ext_vector_type(2))) float v2f;
typedef __attribute__((ext_vector_type(8))) float v8f;

#define CANVAS 128
#define NPTS   17

// One wave: 64 pixels x 16 segments of one path, via V_WMMA_F32_16X16X4_F32.
//   Matmul 1 (pmv2): A1[s] = (-2vx, -2vy, |v|^2, 1)        -> |p - v|^2
//   Matmul 2 (D2)  : A2[s] = (-2wvx, -2wvy, 2(v.wv), 0)    -> -2 (p - v).wv
//   Shared B[:,n]  = (x, y, 1, x^2+y^2) per pixel n of a 16-pixel tile.
// The -2 factor is folded into A2 so the elementwise step needs no extra
// multiply:  t = clamp(D2 * (-0.5/(d^2+eps)));  dist^2 = fma(t, fma(t, d^2, D2), pmv2).
// D layout: lane = pixel (N = lane%16), VGPR r = segment (M = r | r+8 per
// half-wave). Min over segments: in-lane v_min/min3 + shfl_xor(16) + min.
// Tiles processed in pairs so sqrt/sigmoid/store run once per pixel across
// all 32 lanes (full-width coalesced stores).

__device__ __forceinline__ float tile_min_d2(int tilePix, bool lo, int lane,
                                             v2f a1, v2f a2,
                                             const float* d2r,
                                             const float* rcpn)
{
    const int   n   = lane & 15;
    const int   pix = tilePix + n;
    const float x   = (float)(pix & (CANVAS - 1));
    const float y   = (float)(pix >> 7);

    v2f b;
    b.x = lo ? x : 1.0f;
    b.y = lo ? y : fmaf(x, x, y * y);

    v8f zeroc = {0.0f, 0.0f, 0.0f, 0.0f, 0.0f, 0.0f, 0.0f, 0.0f};
    v8f D2 = __builtin_amdgcn_wmma_f32_16x16x4_f32(          // -2*(p-v).wv
        false, a2, false, b, (short)0, zeroc, false, false);
    v8f pmv2 = __builtin_amdgcn_wmma_f32_16x16x4_f32(        // |p-v|^2
        false, a1, false, b, (short)0, zeroc, false, false);

    float m = 3.402823e38f;
#pragma unroll
    for (int r = 0; r < 8; ++r) {
        float d2v = D2[r];
        float t   = __builtin_amdgcn_fmed3f(d2v * rcpn[r], 0.0f, 1.0f);
        float dd  = fmaf(t, fmaf(t, d2r[r], d2v), pmv2[r]);
        m = fminf(m, dd);
    }
    // combine the two half-wave segment groups (M=0..7 with M=8..15)
    m = fminf(m, __shfl_xor(m, 16, 32));
    return fmaxf(m, 0.0f);
}

__global__ __launch_bounds__(256)
void DiffPathRenderer_kernel(const float* __restrict__ traj,
                             const float* __restrict__ pthick,
                             float* __restrict__ out)
{
    const int  lane = threadIdx.x & 31;
    const int  wid  = threadIdx.x >> 5;
    const int  path = blockIdx.y;
    const bool lo   = lane < 16;

    // ---- per-path segment setup (s = lane % 16, duplicated across halves) ----
    const int s = lane & 15;
    const float* tp = traj + path * (NPTS * 2) + 2 * s;
    const float vx  = tp[0] * (float)CANVAS;
    const float vy  = tp[1] * (float)CANVAS;
    const float wx  = tp[2] * (float)CANVAS;
    const float wy  = tp[3] * (float)CANVAS;
    const float wvx = wx - vx, wvy = wy - vy;
    const float d2  = wvx * wvx + wvy * wvy;              // |w - v|^2

    v2f a1, a2;
    a1.x = lo ? (-2.0f * vx) : fmaf(vx, vx, vy * vy);     // K0 | K2
    a1.y = lo ? (-2.0f * vy) : 1.0f;                      // K1 | K3
    a2.x = lo ? (-2.0f * wvx) : (2.0f * fmaf(vx, wvx, vy * wvy));
    a2.y = lo ? (-2.0f * wvy) : 0.0f;

    // per-segment constants for this lane's 8 accumulator rows (hoisted)
    const int segBase = lo ? 0 : 8;
    float d2r[8], rcpn[8];
#pragma unroll
    for (int r = 0; r < 8; ++r) {
        d2r[r]  = __shfl(d2, segBase + r, 16);
        rcpn[r] = -0.5f * __builtin_amdgcn_rcpf(d2r[r] + 1e-5f);
    }

    const float radius = 0.5f * pthick[0];
    const float invrad = __builtin_amdgcn_rcpf(radius);

    // ---- 4 tiles (2 pairs) of 16 pixels per wave ----
    const int waveBase = (blockIdx.x * 8 + wid) * 64;
#pragma unroll
    for (int pair = 0; pair < 2; ++pair) {
        const int base = waveBase + pair * 32;
        float mA = tile_min_d2(base,      lo, lane, a1, a2, d2r, rcpn);
        float mB = tile_min_d2(base + 16, lo, lane, a1, a2, d2r, rcpn);

        // lanes 0-15 carry tile A (pixels base+lane), lanes 16-31 tile B
        // (pixels base+16+(lane-16) = base+lane): one coalesced 32-lane store.
        float m    = lo ? mA : mB;
        float dist = __builtin_amdgcn_sqrtf(m);
        float dk   = __builtin_amdgcn_fmed3f((radius - dist) * invrad, 0.0f, 1.0f);
        float e    = __expf(-70.0f * (dk - 0.9f));
        float sg   = __builtin_amdgcn_rcpf(1.0f + e);

        out[path * (CANVAS * CANVAS) + base + lane] = sg;
    }
}

extern "C" void kernel_launch(void* const* d_in, const int* in_sizes, int n_in,
                              void* d_out, int out_size, void* d_ws, size_t ws_size,
                              hipStream_t stream) {
    (void)in_sizes; (void)n_in; (void)d_ws; (void)ws_size; (void)out_size;
    const float* traj   = (const float*)d_in[0];   // (256, 17, 2) f32, [0,1]
    const float* pthick = (const float*)d_in[1];   // scalar f32
    float* out = (float*)d_out;                    // (256, 128, 128) f32

    // 8 waves/block * 64 pixels/wave = 512 pixels/block; 32 x 256 blocks
    dim3 grid(CANVAS * CANVAS / 512, 256, 1);
    dim3 block(256, 1, 1);
    DiffPathRenderer_kernel<<<grid, block, 0, stream>>>(traj, pthick, out);
}